// ANN_LeafRiver_Sigmoid_qsim_80693845557570
// MI455X (gfx1250) — compile-verified
//
#include <hip/hip_runtime.h>
#include <hip/hip_bf16.h>

typedef __attribute__((ext_vector_type(2))) float v2f;
typedef __attribute__((ext_vector_type(8))) float v8f;

#define XSTRIDE 64   // T*I = 4*16 floats per b-row of x
#define HDIM    64
#define PF      16   // serial-loop prefetch depth (register ring)

// ---------------------------------------------------------------------------
// Phase 1 (per chunk): z[lr][h] = bias + sum_{i<16} x[(b0+lr)*64 + i] * W[i][h]
// One wave -> 16 rows x 64 cols: 16x V_WMMA_F32_16X16X4_F32 (4 K-steps x 4 N-tiles)
// ---------------------------------------------------------------------------
__global__ __launch_bounds__(256) void gemm_xw_wmma(
    const float* __restrict__ x, const float* __restrict__ w,
    const float* __restrict__ bias, float* __restrict__ z,
    long long b0, int nrows, long long Btot)
{
    int wave = (int)((blockIdx.x * (unsigned)blockDim.x + threadIdx.x) >> 5);
    int lane = threadIdx.x & 31;
    int lrow0 = wave * 16;
    if (lrow0 >= nrows) return;          // wave-uniform exit: EXEC stays all-1s

    int half = lane >> 4;                // 0: lanes 0-15, 1: lanes 16-31
    int l15  = lane & 15;

    // A tiles: lane holds row m=l15; VGPR j of k-step k holds K = 4k + 2*half + j
    long long rrow = b0 + lrow0 + l15;
    if (rrow >= Btot) rrow = Btot - 1;   // clamp (partial tile safety)
    const float* xrow = x + rrow * XSTRIDE + 2 * half;
    v2f a[4];
#pragma unroll
    for (int k = 0; k < 4; ++k) {
        float2 t = *(const float2*)(xrow + 4 * k);
        v2f av = { t.x, t.y };
        a[k] = av;
    }

    // B tiles: VGPR j holds K = 4k + 2*half + j, N = 16t + l15 (4KB, L2-resident)
    v2f bm[4][4];
#pragma unroll
    for (int k = 0; k < 4; ++k) {
#pragma unroll
        for (int t = 0; t < 4; ++t) {
            int krow = 4 * k + 2 * half;
            v2f bv = { w[(krow + 0) * HDIM + 16 * t + l15],
                       w[(krow + 1) * HDIM + 16 * t + l15] };
            bm[k][t] = bv;
        }
    }

    v8f acc[4] = {};                     // 4 N-tiles of 16x16 f32 accumulators
#pragma unroll
    for (int k = 0; k < 4; ++k) {
#pragma unroll
        for (int t = 0; t < 4; ++t) {
            acc[t] = __builtin_amdgcn_wmma_f32_16x16x4_f32(
                false, a[k], false, bm[k][t], (short)0, acc[t], false, false);
        }
    }

    float bs = bias[0];
#pragma unroll
    for (int t = 0; t < 4; ++t) {
#pragma unroll
        for (int v = 0; v < 8; ++v) {
            int M = v + 8 * half;        // C/D layout: lanes 0-15 M=v, 16-31 M=v+8
            int lr = lrow0 + M;
            if (lr < nrows)
                z[(size_t)lr * HDIM + 16 * t + l15] = acc[t][v] + bs;
        }
    }
}

// ---------------------------------------------------------------------------
// Phase 2 (per chunk): single wave32 runs the sequential sigmoid recurrence.
// Lane l owns h = {2l, 2l+1}. 16-deep register prefetch ring hides z latency.
// y_out reduction (64 -> 1) via 5 shfl_xor, off the recurrence critical path.
// Recurrent state (y_hs, y_h) carried chunk-to-chunk in d_ws.
// ---------------------------------------------------------------------------
__global__ __launch_bounds__(32) void serial_scan(
    const float* __restrict__ z, const float* __restrict__ wy,
    const float* __restrict__ wln, const float* __restrict__ blnp,
    float* __restrict__ outc, float* __restrict__ state,
    float* __restrict__ tail, int nb, int first, int last)
{
    int lane = threadIdx.x;
    int h0 = lane * 2;

    float wyx = wy[h0],  wyy = wy[h0 + 1];
    float wlx = wln[h0], wly = wln[h0 + 1];
    float bln = blnp[0];

    float hsx, hsy, yhx, yhy;
    if (first) { hsx = 0.f; hsy = 0.f; yhx = 0.f; yhy = 0.f; }
    else {
        hsx = state[h0];        hsy = state[h0 + 1];
        yhx = state[HDIM + h0]; yhy = state[HDIM + h0 + 1];
    }

    const float2* zrow = (const float2*)z + lane;   // step stride = 32 float2
    float2 buf[PF];
#pragma unroll
    for (int j = 0; j < PF; ++j)
        buf[j] = (j < nb) ? zrow[(size_t)j * 32] : make_float2(0.f, 0.f);

    for (int bb = 0; bb < nb; bb += PF) {
#pragma unroll
        for (int j = 0; j < PF; ++j) {
            int b = bb + j;
            if (b >= nb) break;
            float2 zv = buf[j];
            int bp = b + PF;
            if (bp < nb) buf[j] = zrow[(size_t)bp * 32];   // prefetch ahead

            // critical path: fma -> exp -> rcp
            yhx = fmaf(hsx, wyx, zv.x);
            yhy = fmaf(hsy, wyy, zv.y);
            hsx = 1.0f / (1.0f + __expf(-yhx));
            hsy = 1.0f / (1.0f + __expf(-yhy));

            // off-path output reduction across 64 h-values (2/lane, wave32)
            float p = fmaf(hsx, wlx, hsy * wly);
            p += __shfl_xor(p, 16, 32);
            p += __shfl_xor(p, 8, 32);
            p += __shfl_xor(p, 4, 32);
            p += __shfl_xor(p, 2, 32);
            p += __shfl_xor(p, 1, 32);
            if (lane == 0) outc[b] = bln + p;
        }
    }

    state[h0] = hsx;        state[h0 + 1] = hsy;
    state[HDIM + h0] = yhx; state[HDIM + h0 + 1] = yhy;
    if (last) {
        tail[h0] = yhx;        tail[h0 + 1] = yhy;   // y_h_f
        tail[HDIM + h0] = hsx; tail[HDIM + h0 + 1] = hsy;   // y_hs_f
    }
}

// ---------------------------------------------------------------------------
extern "C" void kernel_launch(void* const* d_in, const int* in_sizes, int n_in,
                              void* d_out, int out_size, void* d_ws, size_t ws_size,
                              hipStream_t stream) {
    const float* x    = (const float*)d_in[0];   // (B, 4, 16)
    const float* w    = (const float*)d_in[1];   // (16, 64)
    const float* wy   = (const float*)d_in[2];   // (1, 64)
    const float* bias = (const float*)d_in[3];   // (1,)
    const float* wln  = (const float*)d_in[4];   // (64, 1)
    const float* bln  = (const float*)d_in[5];   // (1,)
    float* out = (float*)d_out;

    long long B = (long long)in_sizes[0] / XSTRIDE;

    // chunk size: multiple of 16 rows, z slab must fit in workspace
    long long Bc = 50000;
    if (Bc > B) Bc = ((B + 15) / 16) * 16;
    while ((size_t)(Bc * HDIM * sizeof(float)) + 512 > ws_size && Bc > 16)
        Bc = (Bc / 2) & ~15LL;

    float* z     = (float*)d_ws;
    float* state = (float*)((char*)d_ws + (size_t)Bc * HDIM * sizeof(float));
    float* tail  = out + B;                      // y_h_f then y_hs_f

    for (long long b0 = 0; b0 < B; b0 += Bc) {
        long long rem = B - b0;
        int nb = (int)(rem < Bc ? rem : Bc);
        int nwaves  = (nb + 15) / 16;
        int nblocks = (nwaves + 7) / 8;          // 8 waves / 256-thread block
        gemm_xw_wmma<<<nblocks, 256, 0, stream>>>(x, w, bias, z, b0, nb, B);
        serial_scan<<<1, 32, 0, stream>>>(z, wy, wln, bln, out + b0, state, tail,
                                          nb, b0 == 0 ? 1 : 0,
                                          (b0 + nb >= B) ? 1 : 0);
    }
}